// ShapePredictor_82858509074815
// MI455X (gfx1250) — compile-verified
//
#include <hip/hip_runtime.h>
#include <hip/hip_bf16.h>

typedef __attribute__((ext_vector_type(2))) float v2f;
typedef __attribute__((ext_vector_type(8))) float v8f;

__constant__ float kPrimes[30] = {2,3,5,7,11,13,17,19,23,29,31,37,41,43,47,
                                  53,59,61,67,71,73,79,83,89,97,101,103,107,109,113};

// ============================================================================
// WMMA f32 GEMM:  C(MxN) = A(MxK) @ B + bias  (optional relu)
//  TRANSB=true : B given as W (N x K) row-major -> B[k][n] = W[n*K+k]
//  TRANSB=false: B given as (K x N) row-major
// Block: 256 threads = 8 waves. Block tile 128x64, K-chunk 32 via LDS.
// Wave w owns rows 16w..16w+15 and all 64 columns: 4 WMMA accumulators.
// A frags: one aligned b64 LDS read (row stride 36 -> 8B/16B alignment).
// B frags: LDS holds k-pairs as v2f, so a frag is one aligned b64 read.
// Requires: N % 64 == 0, K % 32 == 0 (true at all call sites). M guarded.
// ============================================================================
template<bool TRANSB, bool RELU>
__global__ __launch_bounds__(256, 2)
void gemm_kernel(const float* __restrict__ A, long long strideA,
                 const float* __restrict__ B, long long strideB,
                 const float* __restrict__ bias,
                 float* __restrict__ C, long long strideC,
                 int M, int N, int K)
{
    __shared__ float As[128][36];   // [m][k], stride 36: b128 stores, b64 frag loads
    __shared__ v2f   Bs[16][65];    // [k-pair][n] = (B[2k][n], B[2k+1][n])
    const int z = blockIdx.z;
    A += strideA * z; B += strideB * z; C += strideC * z;

    const int m0   = blockIdx.y * 128;
    const int n0   = blockIdx.x * 64;
    const int tid  = threadIdx.x;
    const int lane = tid & 31;
    const int wid  = tid >> 5;
    const int half = lane >> 4;      // 0: lanes 0-15, 1: lanes 16-31
    const int l15  = lane & 15;
    const int aRow = wid * 16 + l15;

    v8f c0 = {0,0,0,0,0,0,0,0};
    v8f c1 = {0,0,0,0,0,0,0,0};
    v8f c2 = {0,0,0,0,0,0,0,0};
    v8f c3 = {0,0,0,0,0,0,0,0};

    for (int k0 = 0; k0 < K; k0 += 32) {
        // ---- prefetch next K-chunk (global_prefetch_b8, speculative) ----
        if (k0 + 32 < K) {
            int pr = tid >> 1, pc = k0 + 32 + (tid & 1) * 16;
            if (m0 + pr < M) __builtin_prefetch(&A[(long long)(m0 + pr) * K + pc], 0, 3);
            if (tid < 64) {
                if (TRANSB) __builtin_prefetch(&B[(long long)(n0 + tid) * K + k0 + 32], 0, 3);
                else        __builtin_prefetch(&B[(long long)(k0 + 32 + (tid >> 1)) * N + n0 + (tid & 1) * 32], 0, 3);
            }
        }
        // ---- A: 128x32 chunk, float4 global loads, float4 LDS stores ----
        #pragma unroll
        for (int it = 0; it < 4; ++it) {
            int idx = it * 256 + tid;            // 1024 float4s
            int r = idx >> 3, cc = (idx & 7) * 4;
            int gr = m0 + r;
            float4 v = (gr < M) ? *(const float4*)&A[(long long)gr * K + (k0 + cc)]
                                : make_float4(0.f, 0.f, 0.f, 0.f);
            *(float4*)&As[r][cc] = v;
        }
        // ---- B: 32(k) x 64(n) chunk into k-pair layout ----
        if (TRANSB) {
            #pragma unroll
            for (int it = 0; it < 2; ++it) {     // 512 float4s along K
                int idx = it * 256 + tid;
                int n = idx >> 3, cc = (idx & 7) * 4;   // k offset, even
                float4 v = *(const float4*)&B[(long long)(n0 + n) * K + (k0 + cc)];
                v2f p0; p0[0] = v.x; p0[1] = v.y;
                v2f p1; p1[0] = v.z; p1[1] = v.w;
                Bs[(cc >> 1) + 0][n] = p0;
                Bs[(cc >> 1) + 1][n] = p1;
            }
        } else {
            #pragma unroll
            for (int it = 0; it < 2; ++it) {     // 512 float4s along N
                int idx = it * 256 + tid;
                int r = idx >> 4, cc = (idx & 15) * 4;  // r = k, cc = n offset
                float4 v = *(const float4*)&B[(long long)(k0 + r) * N + (n0 + cc)];
                Bs[r >> 1][cc + 0][r & 1] = v.x;
                Bs[r >> 1][cc + 1][r & 1] = v.y;
                Bs[r >> 1][cc + 2][r & 1] = v.z;
                Bs[r >> 1][cc + 3][r & 1] = v.w;
            }
        }
        __syncthreads();
        #pragma unroll
        for (int kk = 0; kk < 32; kk += 4) {
            const int kp = (kk >> 1) + half;     // k-pair index for k = kk + 2*half
            const v2f a  = *(const v2f*)&As[aRow][kk + 2 * half];
            const v2f b0 = Bs[kp][l15];
            const v2f b1 = Bs[kp][l15 + 16];
            const v2f b2 = Bs[kp][l15 + 32];
            const v2f b3 = Bs[kp][l15 + 48];
            c0 = __builtin_amdgcn_wmma_f32_16x16x4_f32(false, a, false, b0, (short)0, c0, false, false);
            c1 = __builtin_amdgcn_wmma_f32_16x16x4_f32(false, a, false, b1, (short)0, c1, false, false);
            c2 = __builtin_amdgcn_wmma_f32_16x16x4_f32(false, a, false, b2, (short)0, c2, false, false);
            c3 = __builtin_amdgcn_wmma_f32_16x16x4_f32(false, a, false, b3, (short)0, c3, false, false);
        }
        __syncthreads();
    }

    const int rbase = m0 + wid * 16 + 8 * half;  // C VGPR r -> row r + 8*half
    const int colb  = n0 + l15;
    float bb0 = 0.f, bb1 = 0.f, bb2 = 0.f, bb3 = 0.f;
    if (bias) { bb0 = bias[colb]; bb1 = bias[colb + 16]; bb2 = bias[colb + 32]; bb3 = bias[colb + 48]; }
    #pragma unroll
    for (int r = 0; r < 8; ++r) {
        int row = rbase + r;
        if (row < M) {
            float v0 = c0[r] + bb0, v1 = c1[r] + bb1, v2 = c2[r] + bb2, v3 = c3[r] + bb3;
            if (RELU) {
                v0 = fmaxf(v0, 0.f); v1 = fmaxf(v1, 0.f);
                v2 = fmaxf(v2, 0.f); v3 = fmaxf(v3, 0.f);
            }
            float* cr = C + (long long)row * N + colb;
            cr[0] = v0; cr[16] = v1; cr[32] = v2; cr[48] = v3;
        }
    }
}

// ============================================================================
// Segment mixing weights: rows 0-31 softmax over s (/S), row 32 = 0.2/S, 33-47 = 0
// ============================================================================
__global__ __launch_bounds__(256)
void wseg_kernel(float* __restrict__ W)
{
    const int i = blockIdx.x, tid = threadIdx.x;
    __shared__ float red[256];
    if (i >= 33) { for (int s = tid; s < 2048; s += 256) W[(long long)i*2048 + s] = 0.0f; return; }
    if (i == 32) { for (int s = tid; s < 2048; s += 256) W[(long long)i*2048 + s] = 0.2f/2048.0f; return; }
    const float fi = (float)i;
    const float fa = fi*0.7f + 1.3f, fb = fi*0.5f + 0.8f, fc = sinf(fi*1.7f)*3.0f;
    float loc[8]; float mx = -1e30f;
    #pragma unroll
    for (int j = 0; j < 8; ++j) {
        float s = (float)(tid + j*256);
        float l = sinf(s*fa)*2.0f + cosf(s*fb)*1.5f + fc;
        loc[j] = l; mx = fmaxf(mx, l);
    }
    red[tid] = mx; __syncthreads();
    for (int st = 128; st > 0; st >>= 1) { if (tid < st) red[tid] = fmaxf(red[tid], red[tid+st]); __syncthreads(); }
    mx = red[0]; __syncthreads();
    float sum = 0.0f;
    #pragma unroll
    for (int j = 0; j < 8; ++j) { loc[j] = expf(loc[j] - mx); sum += loc[j]; }
    red[tid] = sum; __syncthreads();
    for (int st = 128; st > 0; st >>= 1) { if (tid < st) red[tid] += red[tid+st]; __syncthreads(); }
    const float inv = 1.0f / (red[0] * 2048.0f);
    #pragma unroll
    for (int j = 0; j < 8; ++j) W[(long long)i*2048 + tid + j*256] = loc[j] * inv;
}

// queries + contiguous seg_mem copy
__global__ __launch_bounds__(256)
void queries_kernel(const float* __restrict__ SEG, const float* __restrict__ pos_emb,
                    float* __restrict__ X, float* __restrict__ SEGC)
{
    const int idx = blockIdx.x * 256 + threadIdx.x;   // 1024*512
    const int dd = idx & 511, t = idx >> 9;
    const int b = t >> 5, i = t & 31;
    const float m = SEG[((long long)b*48 + i)*512 + dd];
    SEGC[idx] = m;
    const float p  = kPrimes[i % 30];
    const float fi = (float)i;
    const float fp = sinf(m*p*0.1f) + cosf(m*p*0.07f)*0.8f + tanhf(m*p*0.13f)*0.6f;
    const float ie = sinf(fi*2.1f)*m*0.7f + cosf(fi*3.3f)*m*0.5f + sinf(fi*5.7f)*0.4f;
    X[idx] = fp + ie + 0.05f * pos_emb[(long long)i*512 + dd];
}

__global__ __launch_bounds__(256)
void broader_kernel(const float* __restrict__ SEG, float* __restrict__ BRD)
{
    const int idx = blockIdx.x * 256 + threadIdx.x;   // 32*512
    const int dd = idx & 511, b = idx >> 9;
    BRD[idx] = SEG[((long long)b*48 + 32)*512 + dd];
}

// sf = feat + stop_pos[i]
__global__ __launch_bounds__(256)
void sf_kernel(const float* __restrict__ X, const float* __restrict__ stop_pos,
               float* __restrict__ SF)
{
    const int idx = blockIdx.x * 256 + threadIdx.x;   // 1024*512
    const int dd = idx & 511, t = idx >> 9, i = t & 31;
    SF[idx] = X[idx] + stop_pos[(long long)i*512 + dd];
}

// X = LayerNorm(X + R)  (R may be null); one block per 512-wide row
__global__ __launch_bounds__(256)
void addln_kernel(float* __restrict__ X, const float* __restrict__ R,
                  const float* __restrict__ g, const float* __restrict__ b)
{
    const int row = blockIdx.x, tid = threadIdx.x;
    __shared__ float red[256];
    const long long base = (long long)row * 512;
    float v0 = X[base + tid], v1 = X[base + tid + 256];
    if (R) { v0 += R[base + tid]; v1 += R[base + tid + 256]; }
    red[tid] = v0 + v1; __syncthreads();
    for (int st = 128; st > 0; st >>= 1) { if (tid < st) red[tid] += red[tid+st]; __syncthreads(); }
    const float mu = red[0] * (1.0f/512.0f); __syncthreads();
    const float d0 = v0 - mu, d1 = v1 - mu;
    red[tid] = d0*d0 + d1*d1; __syncthreads();
    for (int st = 128; st > 0; st >>= 1) { if (tid < st) red[tid] += red[tid+st]; __syncthreads(); }
    const float inv = rsqrtf(red[0] * (1.0f/512.0f) + 1e-5f);
    X[base + tid]       = d0 * inv * g[tid]       + b[tid];
    X[base + tid + 256] = d1 * inv * g[tid + 256] + b[tid + 256];
}

// 2-key cross attention; one wave per (token, head), float2 lanes + shfl reduce
__global__ __launch_bounds__(256)
void attn_kernel(const float* __restrict__ Q,  const float* __restrict__ KS,
                 const float* __restrict__ KB, const float* __restrict__ VS,
                 const float* __restrict__ VB, float* __restrict__ O)
{
    const int t = blockIdx.x;            // 1024 tokens
    const int h = threadIdx.x >> 5;      // head 0..7
    const int lane = threadIdx.x & 31;
    const int b = t >> 5;
    const long long toff = (long long)t*512 + h*64 + lane*2;
    const long long boff = (long long)b*512 + h*64 + lane*2;
    const float2 qv  = *(const float2*)(Q  + toff);
    const float2 k0v = *(const float2*)(KS + toff);
    const float2 k1v = *(const float2*)(KB + boff);
    float d0 = qv.x*k0v.x + qv.y*k0v.y;
    float d1 = qv.x*k1v.x + qv.y*k1v.y;
    #pragma unroll
    for (int m = 16; m > 0; m >>= 1) {
        d0 += __shfl_xor(d0, m, 32);
        d1 += __shfl_xor(d1, m, 32);
    }
    d0 *= 0.125f; d1 *= 0.125f;          // 1/sqrt(64)
    const float mx = fmaxf(d0, d1);
    const float e0 = expf(d0 - mx), e1 = expf(d1 - mx);
    const float inv = 1.0f / (e0 + e1);
    const float a0 = e0 * inv, a1 = e1 * inv;
    const float2 v0 = *(const float2*)(VS + toff);
    const float2 v1 = *(const float2*)(VB + boff);
    float2 o; o.x = a0*v0.x + a1*v1.x; o.y = a0*v0.y + a1*v1.y;
    *(float2*)(O + toff) = o;
}

// Output heads; stop hidden layer comes in precomputed via WMMA GEMM (STH)
__global__ __launch_bounds__(256)
void heads_kernel(const float* __restrict__ FEAT, const float* __restrict__ HC2,
                  const float* __restrict__ BRD,  const float* __restrict__ STH,
                  const float* __restrict__ ch_w3, const float* __restrict__ ch_b3,
                  const float* __restrict__ th_w,  const float* __restrict__ th_b,
                  const float* __restrict__ sh_w2, const float* __restrict__ sh_b2,
                  float* __restrict__ out)
{
    const int t = blockIdx.x, tid = threadIdx.x;
    const int b = t >> 5, i = t & 31;
    __shared__ float feat_s[512], hc2_s[512], red[256];
    __shared__ float cvals[6], tvals[3], sumH_s, stop_s;
    for (int j = tid; j < 512; j += 256) {
        feat_s[j] = FEAT[(long long)t*512 + j];
        hc2_s[j]  = HC2[(long long)t*512 + j];
    }
    red[tid] = BRD[(long long)b*512 + tid] + BRD[(long long)b*512 + tid + 256];
    __syncthreads();
    for (int st = 128; st > 0; st >>= 1) { if (tid < st) red[tid] += red[tid+st]; __syncthreads(); }
    if (tid == 0) sumH_s = red[0] * 5.0f;   // broader = mean*0.2 -> mean.sum = sum/0.2
    __syncthreads();
    red[tid] = STH[(long long)t*256 + tid] * sh_w2[tid];
    __syncthreads();
    for (int st = 128; st > 0; st >>= 1) { if (tid < st) red[tid] += red[tid+st]; __syncthreads(); }
    if (tid == 0) stop_s = red[0] + sh_b2[0];
    if (tid < 6) {
        float s = 0.0f; const float* w = ch_w3 + (long long)tid * 512;
        for (int j = 0; j < 512; ++j) s += hc2_s[j] * w[j];
        cvals[tid] = s + ch_b3[tid];
    }
    if (tid >= 8 && tid < 11) {
        const int jj = tid - 8;
        float s = 0.0f; const float* w = th_w + (long long)jj * 512;
        for (int j = 0; j < 512; ++j) s += feat_s[j] * w[j];
        tvals[jj] = s + th_b[jj];
    }
    __syncthreads();
    if (tid == 0) {
        const float fi = (float)i;
        const float scale  = 0.8f + 0.4f * sinf(fi * 1.7f);
        const float offset = 0.1f * cosf(fi * 2.3f);
        const float scalar = sinf(sumH_s * (fi + 1.0f) * 0.3f);
        for (int j = 0; j < 6; ++j) {
            const float v = tanhf(cvals[j]) * scale + offset + scalar * 0.15f;
            out[(long long)t*6 + j] = 1.0f / (1.0f + expf(-v));
        }
        float* tl = out + 6144 + (long long)t*3;
        tl[0] = tvals[0] + 0.3f * sinf(fi * 2.1f);
        tl[1] = tvals[1] + 0.3f * cosf(fi * 1.8f);
        tl[2] = tvals[2] + 0.3f * sinf(fi * 3.2f);
        out[6144 + 3072 + t] = stop_s + (-2.0f + fi * 0.4f);
    }
}

// ============================================================================
static inline void launch_gemm(hipStream_t s,
                               const float* A, long long sA,
                               const float* B, long long sB,
                               const float* bias,
                               float* C, long long sC,
                               int M, int N, int K, int batches,
                               bool transB, bool relu)
{
    dim3 grid(N / 64, (M + 127) / 128, batches);
    dim3 block(256);
    if (transB) {
        if (relu) gemm_kernel<true , true ><<<grid, block, 0, s>>>(A, sA, B, sB, bias, C, sC, M, N, K);
        else      gemm_kernel<true , false><<<grid, block, 0, s>>>(A, sA, B, sB, bias, C, sC, M, N, K);
    } else {
        if (relu) gemm_kernel<false, true ><<<grid, block, 0, s>>>(A, sA, B, sB, bias, C, sC, M, N, K);
        else      gemm_kernel<false, false><<<grid, block, 0, s>>>(A, sA, B, sB, bias, C, sC, M, N, K);
    }
}

// workspace layout (floats)
#define OFF_WSEG 0LL                               // 48*2048
#define OFF_SEG  (OFF_WSEG + 48LL*2048)            // 32*48*512
#define OFF_SEGC (OFF_SEG  + 32LL*48*512)          // 1024*512
#define OFF_BRD  (OFF_SEGC + 1024LL*512)           // 32*512
#define OFF_X    (OFF_BRD  + 32LL*512)             // 1024*512
#define OFF_T1   (OFF_X    + 1024LL*512)           // 1024*2048
#define OFF_T2   (OFF_T1   + 1024LL*2048)          // 1024*512
#define OFF_KS   (OFF_T2   + 1024LL*512)           // 1024*512 (also stop hidden)
#define OFF_VS   (OFF_KS   + 1024LL*512)           // 1024*512
#define OFF_KB   (OFF_VS   + 1024LL*512)           // 32*512
#define OFF_VB   (OFF_KB   + 32LL*512)             // 32*512
#define OFF_ATT  (OFF_VB   + 32LL*512)             // 1024*512 (also sf buffer)

extern "C" void kernel_launch(void* const* d_in, const int* in_sizes, int n_in,
                              void* d_out, int out_size, void* d_ws, size_t ws_size,
                              hipStream_t stream)
{
    (void)in_sizes; (void)n_in; (void)out_size; (void)ws_size;
    const float* H        = (const float*)d_in[0];
    const float* pos_emb  = (const float*)d_in[1];
    const float* sa_in_w  = (const float*)d_in[2];
    const float* sa_in_b  = (const float*)d_in[3];
    const float* sa_out_w = (const float*)d_in[4];
    const float* sa_out_b = (const float*)d_in[5];
    const float* ca_in_w  = (const float*)d_in[6];
    const float* ca_in_b  = (const float*)d_in[7];
    const float* ca_out_w = (const float*)d_in[8];
    const float* ca_out_b = (const float*)d_in[9];
    const float* ln1_g = (const float*)d_in[10];
    const float* ln1_b = (const float*)d_in[11];
    const float* ln2_g = (const float*)d_in[12];
    const float* ln2_b = (const float*)d_in[13];
    const float* ln3_g = (const float*)d_in[14];
    const float* ln3_b = (const float*)d_in[15];
    const float* ffn_w1 = (const float*)d_in[16];
    const float* ffn_b1 = (const float*)d_in[17];
    const float* ffn_w2 = (const float*)d_in[18];
    const float* ffn_b2 = (const float*)d_in[19];
    const float* fn_g  = (const float*)d_in[20];
    const float* fn_b  = (const float*)d_in[21];
    const float* ch_w1 = (const float*)d_in[22];
    const float* ch_b1 = (const float*)d_in[23];
    const float* ch_w2 = (const float*)d_in[24];
    const float* ch_b2 = (const float*)d_in[25];
    const float* ch_w3 = (const float*)d_in[26];
    const float* ch_b3 = (const float*)d_in[27];
    const float* th_w  = (const float*)d_in[28];
    const float* th_b  = (const float*)d_in[29];
    const float* sh_w1 = (const float*)d_in[30];
    const float* sh_b1 = (const float*)d_in[31];
    const float* sh_w2 = (const float*)d_in[32];
    const float* sh_b2 = (const float*)d_in[33];
    const float* stop_pos = (const float*)d_in[34];

    float* ws   = (float*)d_ws;
    float* WSEG = ws + OFF_WSEG;
    float* SEG  = ws + OFF_SEG;
    float* SEGC = ws + OFF_SEGC;
    float* BRD  = ws + OFF_BRD;
    float* X    = ws + OFF_X;
    float* T1   = ws + OFF_T1;
    float* T2   = ws + OFF_T2;
    float* KS   = ws + OFF_KS;
    float* VS   = ws + OFF_VS;
    float* KB   = ws + OFF_KB;
    float* VB   = ws + OFF_VB;
    float* ATT  = ws + OFF_ATT;

    // 1) deterministic mixing weights, 2) batched WMMA GEMM over H_memory (HBM pass)
    wseg_kernel<<<48, 256, 0, stream>>>(WSEG);
    launch_gemm(stream, WSEG, 0LL, H, 2048LL*512, nullptr, SEG, 48LL*512,
                48, 512, 2048, 32, /*transB=*/false, /*relu=*/false);
    queries_kernel<<<2048, 256, 0, stream>>>(SEG, pos_emb, X, SEGC);
    broader_kernel<<<64, 256, 0, stream>>>(SEG, BRD);

    for (int l = 0; l < 4; ++l) {
        const float* saw  = sa_in_w  + (long long)l*1536*512;
        const float* sab  = sa_in_b  + (long long)l*1536;
        const float* sow  = sa_out_w + (long long)l*512*512;
        const float* sob  = sa_out_b + (long long)l*512;
        const float* caw  = ca_in_w  + (long long)l*1536*512;
        const float* cab  = ca_in_b  + (long long)l*1536;
        const float* cow  = ca_out_w + (long long)l*512*512;
        const float* cob  = ca_out_b + (long long)l*512;

        // self-attn with Tq=Tk=1: softmax == 1 -> out = OutProj(Vproj(x))
        launch_gemm(stream, X,  0, saw + 1024LL*512, 0, sab + 1024, T1, 0, 1024, 512, 512, 1, true, false);
        launch_gemm(stream, T1, 0, sow,              0, sob,        T2, 0, 1024, 512, 512, 1, true, false);
        addln_kernel<<<1024, 256, 0, stream>>>(X, T2, ln1_g + l*512, ln1_b + l*512);

        // cross-attn, Tk=2 (seg row + broader row)
        launch_gemm(stream, X,    0, caw,              0, cab,        T1, 0, 1024, 512, 512, 1, true, false); // Q
        launch_gemm(stream, SEGC, 0, caw + 512LL*512,  0, cab + 512,  KS, 0, 1024, 512, 512, 1, true, false); // K seg
        launch_gemm(stream, SEGC, 0, caw + 1024LL*512, 0, cab + 1024, VS, 0, 1024, 512, 512, 1, true, false); // V seg
        launch_gemm(stream, BRD,  0, caw + 512LL*512,  0, cab + 512,  KB, 0,   32, 512, 512, 1, true, false); // K brd
        launch_gemm(stream, BRD,  0, caw + 1024LL*512, 0, cab + 1024, VB, 0,   32, 512, 512, 1, true, false); // V brd
        attn_kernel<<<1024, 256, 0, stream>>>(T1, KS, KB, VS, VB, ATT);
        launch_gemm(stream, ATT, 0, cow, 0, cob, T2, 0, 1024, 512, 512, 1, true, false);
        addln_kernel<<<1024, 256, 0, stream>>>(X, T2, ln2_g + l*512, ln2_b + l*512);

        // FFN 512 -> 2048 -> 512
        launch_gemm(stream, X,  0, ffn_w1 + (long long)l*2048*512, 0, ffn_b1 + (long long)l*2048,
                    T1, 0, 1024, 2048, 512, 1, true, true);
        launch_gemm(stream, T1, 0, ffn_w2 + (long long)l*512*2048, 0, ffn_b2 + (long long)l*512,
                    T2, 0, 1024, 512, 2048, 1, true, false);
        addln_kernel<<<1024, 256, 0, stream>>>(X, T2, ln3_g + l*512, ln3_b + l*512);
    }

    // final LN, stop-hidden GEMM, coordinate-head MLP, fused output heads
    addln_kernel<<<1024, 256, 0, stream>>>(X, nullptr, fn_g, fn_b);
    sf_kernel<<<2048, 256, 0, stream>>>(X, stop_pos, ATT);                 // sf = feat + stop_pos
    launch_gemm(stream, ATT, 0, sh_w1, 0, sh_b1, KS, 0, 1024, 256, 512, 1, true, true);  // stop hidden
    launch_gemm(stream, X,   0, ch_w1, 0, ch_b1, T1, 0, 1024, 1024, 512,  1, true, true);
    launch_gemm(stream, T1,  0, ch_w2, 0, ch_b2, T2, 0, 1024, 512,  1024, 1, true, true);
    heads_kernel<<<1024, 256, 0, stream>>>(X, T2, BRD, KS, ch_w3, ch_b3, th_w, th_b,
                                           sh_w2, sh_b2, (float*)d_out);
}